// MultiHeadSageLayer_66168266162367
// MI455X (gfx1250) — compile-verified
//
#include <hip/hip_runtime.h>
#include <hip/hip_bf16.h>

typedef __attribute__((ext_vector_type(2))) float v2f;
typedef __attribute__((ext_vector_type(8))) float v8f;

#define N_NODES_  100000
#define N_EDGES_  1600000
#define DIM_      128
#define N_HEADS_  4

// ---------------------------------------------------------------------------
// Average the per-head weight matrices: Ws_avg = mean_h W_self[h], etc.
// 16384 elements per matrix; threads < 128 also average the bias.
// ---------------------------------------------------------------------------
__global__ __launch_bounds__(256)
void prep_weights_kernel(const float* __restrict__ Ws_in,
                         const float* __restrict__ Wn_in,
                         const float* __restrict__ b_in,
                         float* __restrict__ Ws,
                         float* __restrict__ Wn,
                         float* __restrict__ bavg) {
    int i = blockIdx.x * blockDim.x + threadIdx.x;
    if (i < DIM_ * DIM_) {
        float s = 0.f, n = 0.f;
#pragma unroll
        for (int hd = 0; hd < N_HEADS_; ++hd) {
            s += Ws_in[hd * DIM_ * DIM_ + i];
            n += Wn_in[hd * DIM_ * DIM_ + i];
        }
        Ws[i] = s * 0.25f;
        Wn[i] = n * 0.25f;
    }
    if (i < DIM_) {
        float s = 0.f;
#pragma unroll
        for (int hd = 0; hd < N_HEADS_; ++hd) s += b_in[hd * DIM_ + i];
        bavg[i] = s * 0.25f;
    }
}

// ---------------------------------------------------------------------------
// Edge scatter: one wave (32 lanes) per edge; each lane moves 4 floats.
// neigh_sum accumulates directly into d_out (zeroed beforehand).
// Lane 0 also bumps the degree counter. unsafeAtomicAdd -> global_atomic_add_f32.
// ---------------------------------------------------------------------------
__global__ __launch_bounds__(256)
void scatter_kernel(const float4* __restrict__ h4,
                    const int* __restrict__ src,
                    const int* __restrict__ dst,
                    float* __restrict__ out,     // neigh_sum accumulator
                    float* __restrict__ deg,
                    int n_edges) {
    int gid  = blockIdx.x * blockDim.x + threadIdx.x;
    int e    = gid >> 5;
    int lane = gid & 31;
    if (e >= n_edges) return;
    int s = src[e];
    int d = dst[e];
    float4 v = h4[(size_t)s * (DIM_ / 4) + lane];
    float* o = out + (size_t)d * DIM_ + lane * 4;
    unsafeAtomicAdd(o + 0, v.x);
    unsafeAtomicAdd(o + 1, v.y);
    unsafeAtomicAdd(o + 2, v.z);
    unsafeAtomicAdd(o + 3, v.w);
    if (lane == 0) unsafeAtomicAdd(deg + d, 1.0f);
}

// ---------------------------------------------------------------------------
// Fused dual GEMM with V_WMMA_F32_16X16X4_F32 (fp32, matches reference math):
//   out[n,:] = h[n,:] @ Ws_avg + (neigh_sum[n,:]/max(deg,1)) @ Wn_avg + b_avg
// Block = 256 threads = 8 waves; block owns 16 rows; wave w owns columns
// [16w,16w+16). neigh_sum lives in `out`, which is overwritten after a
// __syncthreads() once every wave has consumed its loads via the WMMAs.
// ---------------------------------------------------------------------------
__global__ __launch_bounds__(256)
void sage_gemm_kernel(const float* __restrict__ h,
                      const float* __restrict__ Ws,
                      const float* __restrict__ Wn,
                      const float* __restrict__ bavg,
                      const float* __restrict__ deg,
                      float* __restrict__ out) {
    const int wave = threadIdx.x >> 5;   // 0..7 -> column tile
    const int lane = threadIdx.x & 31;
    const int half = lane >> 4;          // 0 or 1
    const int lm   = lane & 15;
    const int col  = wave * 16 + lm;
    const int row  = blockIdx.x * 16 + lm;

    const float rd = 1.0f / fmaxf(deg[row], 1.0f);
    const float* __restrict__ hrow = h   + (size_t)row * DIM_;
    const float* __restrict__ nrow = out + (size_t)row * DIM_;  // neigh_sum

    v8f c = {0.f, 0.f, 0.f, 0.f, 0.f, 0.f, 0.f, 0.f};

    // Self path: c += A(h) * Ws
#pragma unroll 8
    for (int k = 0; k < DIM_; k += 4) {
        const int kk = k + 2 * half;
        v2f a; a.x = hrow[kk];              a.y = hrow[kk + 1];
        v2f b; b.x = Ws[kk * DIM_ + col];   b.y = Ws[(kk + 1) * DIM_ + col];
        c = __builtin_amdgcn_wmma_f32_16x16x4_f32(false, a, false, b,
                                                  (short)0, c, false, false);
    }
    // Neighbor path: c += A(neigh_sum * 1/deg) * Wn
#pragma unroll 8
    for (int k = 0; k < DIM_; k += 4) {
        const int kk = k + 2 * half;
        v2f a; a.x = nrow[kk] * rd;         a.y = nrow[kk + 1] * rd;
        v2f b; b.x = Wn[kk * DIM_ + col];   b.y = Wn[(kk + 1) * DIM_ + col];
        c = __builtin_amdgcn_wmma_f32_16x16x4_f32(false, a, false, b,
                                                  (short)0, c, false, false);
    }

    const float bb = bavg[col];
#pragma unroll
    for (int i = 0; i < 8; ++i) c[i] += bb;

    // Every wave has consumed neigh_sum for all 16 rows before anyone stores.
    __syncthreads();

#pragma unroll
    for (int i = 0; i < 8; ++i)
        out[(size_t)(blockIdx.x * 16 + i + 8 * half) * DIM_ + col] = c[i];
}

// ---------------------------------------------------------------------------
extern "C" void kernel_launch(void* const* d_in, const int* in_sizes, int n_in,
                              void* d_out, int out_size, void* d_ws, size_t ws_size,
                              hipStream_t stream) {
    const float* h      = (const float*)d_in[0];
    const int*   src    = (const int*)d_in[1];
    const int*   dst    = (const int*)d_in[2];
    const float* W_self = (const float*)d_in[3];
    const float* W_neigh= (const float*)d_in[4];
    const float* b      = (const float*)d_in[5];
    float* out = (float*)d_out;

    const int n_nodes = in_sizes[0] / DIM_;   // 100000
    const int n_edges = in_sizes[1];          // 1600000

    // Workspace layout (floats): deg[n_nodes] | Ws_avg[128*128] | Wn_avg[128*128] | b_avg[128]
    float* deg  = (float*)d_ws;
    float* Ws   = deg + n_nodes;
    float* Wn   = Ws + DIM_ * DIM_;
    float* bavg = Wn + DIM_ * DIM_;
    (void)ws_size; (void)n_in; (void)out_size;

    // 1) Zero the neigh_sum accumulator (aliased with d_out) and degrees.
    hipMemsetAsync(out, 0, (size_t)n_nodes * DIM_ * sizeof(float), stream);
    hipMemsetAsync(deg, 0, (size_t)n_nodes * sizeof(float), stream);

    // 2) Average heads into Ws/Wn/b_avg.
    prep_weights_kernel<<<(DIM_ * DIM_ + 255) / 256, 256, 0, stream>>>(
        W_self, W_neigh, b, Ws, Wn, bavg);

    // 3) Edge scatter: one wave per edge.
    {
        long long threads = (long long)n_edges * 32;
        int blocks = (int)((threads + 255) / 256);
        scatter_kernel<<<blocks, 256, 0, stream>>>(
            (const float4*)h, src, dst, out, deg, n_edges);
    }

    // 4) Fused WMMA GEMM, 16 rows per block (100000 / 16 = 6250 blocks).
    sage_gemm_kernel<<<n_nodes / 16, 256, 0, stream>>>(h, Ws, Wn, bavg, deg, out);
}